// S5SSM_40501541601284
// MI455X (gfx1250) — compile-verified
//
#include <hip/hip_runtime.h>

typedef __attribute__((ext_vector_type(2))) float v2f;
typedef __attribute__((ext_vector_type(4))) float v4f;
typedef __attribute__((ext_vector_type(8))) float v8f;
typedef int gv4i __attribute__((vector_size(16)));            // builtin src type
typedef __attribute__((address_space(3))) gv4i lds_gv4i;      // builtin dst type

#define L_DIM   8192
#define H_DIM   512
#define P_DIM   512
#define P2_DIM  1024
#define NCHUNK  128
#define CHUNK   64
#define KC      32            // K-chunk staged in LDS per iteration
#define BN      128           // block N tile (per-wave N tile = BN)
#define NQ      (KC / 4)      // 8 k-steps per chunk
#define NJ      (BN / 16)     // 8 accumulators per wave

#ifdef __has_builtin
#if __has_builtin(__builtin_amdgcn_global_load_async_to_lds_b128)
#define USE_ASYNC_LDS 1
#endif
#endif
#ifndef USE_ASYNC_LDS
#define USE_ASYNC_LDS 0
#endif

__device__ __forceinline__ void wait_async_lds() {
#ifdef __has_builtin
#if __has_builtin(__builtin_amdgcn_s_wait_asynccnt)
  __builtin_amdgcn_s_wait_asynccnt(0);
  return;
#else
  asm volatile("s_wait_asynccnt 0x0" ::: "memory");
  return;
#endif
#else
  asm volatile("s_wait_asynccnt 0x0" ::: "memory");
#endif
}

__device__ __forceinline__ void async_cp_b128(const float* src, float* lds_dst) {
#if USE_ASYNC_LDS
  __builtin_amdgcn_global_load_async_to_lds_b128(
      (gv4i*)(uintptr_t)src, (lds_gv4i*)(uintptr_t)lds_dst, 0, 0);
#endif
}

// ---------------------------------------------------------------------------
// Discretization: Lambda_bar = exp(Lambda*Delta), scale = (Lambda_bar-1)/Lambda
// ---------------------------------------------------------------------------
__global__ __launch_bounds__(P_DIM) void pre_lambda_kernel(
    const float* __restrict__ Lre, const float* __restrict__ Lim,
    const float* __restrict__ logstep,
    float* __restrict__ lbr_o, float* __restrict__ lbi_o,
    float* __restrict__ scr_o, float* __restrict__ sci_o) {
  int p = threadIdx.x;
  float dt = __expf(logstep[p]);
  float lr = Lre[p], li = Lim[p];
  float er = __expf(lr * dt);
  float s, c;
  __sincosf(li * dt, &s, &c);
  float lbr = er * c, lbi = er * s;
  float nr = lbr - 1.0f, ni = lbi;
  float inv = 1.0f / (lr * lr + li * li);
  lbr_o[p] = lbr;
  lbi_o[p] = lbi;
  scr_o[p] = (nr * lr + ni * li) * inv;
  sci_o[p] = (ni * lr - nr * li) * inv;
}

// Bst (P2, H): rows [0,512) = Re(B_bar), rows [512,1024) = Im(B_bar)
__global__ __launch_bounds__(256) void build_B_kernel(
    const float* __restrict__ Bin, const float* __restrict__ scr,
    const float* __restrict__ sci, float* __restrict__ Bst) {
  int idx = blockIdx.x * blockDim.x + threadIdx.x;  // p*512 + h
  int p = idx >> 9, h = idx & 511;
  float br = Bin[2 * idx], bi = Bin[2 * idx + 1];
  float sr = scr[p], si = sci[p];
  Bst[(size_t)p * H_DIM + h]           = sr * br - si * bi;
  Bst[(size_t)(p + P_DIM) * H_DIM + h] = sr * bi + si * br;
}

// Cst (H, P2): cols [0,512) = Re(C), cols [512,1024) = -Im(C)
__global__ __launch_bounds__(256) void build_C_kernel(
    const float* __restrict__ Cin, float* __restrict__ Cst) {
  int idx = blockIdx.x * blockDim.x + threadIdx.x;  // h*512 + p
  int h = idx >> 9, p = idx & 511;
  Cst[(size_t)h * P2_DIM + p]         = Cin[2 * idx];
  Cst[(size_t)h * P2_DIM + p + P_DIM] = -Cin[2 * idx + 1];
}

// ---------------------------------------------------------------------------
// f32 WMMA GEMM:  Out(M,N) = A(M,K) * Bm(N,K)^T   [+ D[n]*U[m,n] if fuse]
// Block = 4 waves = 64(M) x 128(N); wave = 16(M) x 128(N), 8 accumulators.
// B staged in LDS (double-buffered, async global->LDS).
// A fragments register-double-buffered across chunks;
// B fragments register-prefetched one K-step ahead within a chunk.
// LDS B layout: per buffer [kq][n] of float4 (k = 4*kq..4*kq+3 of row n).
// ---------------------------------------------------------------------------
__global__ __launch_bounds__(128) void wmma_gemm_kernel(
    const float* __restrict__ A, const float* __restrict__ Bm,
    float* __restrict__ Out, int K, int N, int fuse,
    const float* __restrict__ D, const float* __restrict__ U) {
  __shared__ float Bsh[2][NQ * BN * 4];  // 2 x 16KB

  const int tid  = threadIdx.x;
  const int lane = tid & 31;
  const int wave = tid >> 5;
  const int hi   = lane >> 4;  // K sub-pair selector
  const int ln   = lane & 15;
  const int m0 = blockIdx.x * 64 + wave * 16;
  const int n0 = blockIdx.y * BN;
  const int NKC = K / KC;

  // A fragment pointer: lane holds row m0+ln, K pair at k + 2*hi
  const float* pa = A + (size_t)(m0 + ln) * K + 2 * hi;

  // B staging map: thread -> (row sn + 16*it, float4 index sj within chunk)
  const int sn = tid >> 3;  // 0..15
  const int sj = tid & 7;   // 0..7
  const float* pbs = Bm + (size_t)(n0 + sn) * K + 4 * sj;

  v8f acc[NJ] = {};
  v2f Ar[NQ], An[NQ];
  v2f bf[2][NJ];

#pragma unroll
  for (int q = 0; q < NQ; ++q) Ar[q] = *(const v2f*)(pa + 4 * q);

#if USE_ASYNC_LDS
  // ---- async global -> LDS pipeline ----
#pragma unroll
  for (int it = 0; it < 8; ++it)
    async_cp_b128(pbs + (size_t)(16 * it) * K,
                  &Bsh[0][(sj * BN + sn + 16 * it) * 4]);
  wait_async_lds();
  __syncthreads();
#else
  {
    v4f breg[8];
#pragma unroll
    for (int it = 0; it < 8; ++it)
      breg[it] = *(const v4f*)(pbs + (size_t)(16 * it) * K);
#pragma unroll
    for (int it = 0; it < 8; ++it)
      *(v4f*)&Bsh[0][(sj * BN + sn + 16 * it) * 4] = breg[it];
    __syncthreads();
  }
#endif

  int cur = 0;
  for (int kc = 0; kc < NKC; ++kc) {
#if !USE_ASYNC_LDS
    v4f breg[8];
#endif
    if (kc + 1 < NKC) {
#if USE_ASYNC_LDS
#pragma unroll
      for (int it = 0; it < 8; ++it)
        async_cp_b128(pbs + (size_t)(16 * it) * K + (kc + 1) * KC,
                      &Bsh[cur ^ 1][(sj * BN + sn + 16 * it) * 4]);
#else
#pragma unroll
      for (int it = 0; it < 8; ++it)
        breg[it] = *(const v4f*)(pbs + (size_t)(16 * it) * K + (kc + 1) * KC);
#endif
#pragma unroll
      for (int q = 0; q < NQ; ++q)
        An[q] = *(const v2f*)(pa + (kc + 1) * KC + 4 * q);
    }

    const float* bl = &Bsh[cur][0];
    // preload B fragments for k-step 0
#pragma unroll
    for (int j = 0; j < NJ; ++j)
      bf[0][j] = *(const v2f*)(bl + (16 * j + ln) * 4 + 2 * hi);

#pragma unroll
    for (int q = 0; q < NQ; ++q) {
      if (q + 1 < NQ) {  // prefetch next k-step's B fragments
#pragma unroll
        for (int j = 0; j < NJ; ++j)
          bf[(q + 1) & 1][j] =
              *(const v2f*)(bl + ((q + 1) * BN + 16 * j + ln) * 4 + 2 * hi);
      }
      v2f a = Ar[q];
#pragma unroll
      for (int j = 0; j < NJ; ++j)
        acc[j] = __builtin_amdgcn_wmma_f32_16x16x4_f32(
            false, a, false, bf[q & 1][j], (short)0, acc[j], false, false);
    }

#if USE_ASYNC_LDS
    wait_async_lds();
    __syncthreads();
#else
    __syncthreads();
    if (kc + 1 < NKC) {
#pragma unroll
      for (int it = 0; it < 8; ++it)
        *(v4f*)&Bsh[cur ^ 1][(sj * BN + sn + 16 * it) * 4] = breg[it];
    }
    __syncthreads();
#endif
#pragma unroll
    for (int q = 0; q < NQ; ++q) Ar[q] = An[q];
    cur ^= 1;
  }

  // C/D layout: vgpr i, lane -> (M = m0 + i + 8*hi, N = n0 + 16*j + ln)
#pragma unroll
  for (int j = 0; j < NJ; ++j) {
#pragma unroll
    for (int i = 0; i < 8; ++i) {
      int m = m0 + i + 8 * hi;
      int n = n0 + 16 * j + ln;
      float v = acc[j][i];
      if (fuse) v += D[n] * U[(size_t)m * N + n];
      Out[(size_t)m * N + n] = v;
    }
  }
}

// ---------------------------------------------------------------------------
// Chunked scan over L (constant diagonal A): x_t = Lb * x_{t-1} + Bu_t
// X layout: (L, P2) with re at col p, im at col p+512.  In-place.
// ---------------------------------------------------------------------------
__global__ __launch_bounds__(P_DIM) void scanA_kernel(
    float* __restrict__ X, const float* __restrict__ lbr_a,
    const float* __restrict__ lbi_a, float* __restrict__ carry) {
  int p = threadIdx.x;
  int c = blockIdx.x;
  float lbr = lbr_a[p], lbi = lbi_a[p];
  float xr = 0.f, xi = 0.f;
  size_t base = (size_t)c * CHUNK * P2_DIM + p;
  for (int j = 0; j < CHUNK; ++j) {
    float br = X[base], bi = X[base + P_DIM];
    float nr = lbr * xr - lbi * xi + br;
    float ni = lbr * xi + lbi * xr + bi;
    X[base] = nr;
    X[base + P_DIM] = ni;
    xr = nr;
    xi = ni;
    base += P2_DIM;
  }
  carry[(size_t)c * P2_DIM + p]         = xr;
  carry[(size_t)c * P2_DIM + p + P_DIM] = xi;
}

__global__ __launch_bounds__(P_DIM) void scanB_kernel(
    const float* __restrict__ carry, float* __restrict__ carry_in,
    const float* __restrict__ lbr_a, const float* __restrict__ lbi_a) {
  int p = threadIdx.x;
  float ar = lbr_a[p], ai = lbi_a[p];
#pragma unroll
  for (int s = 0; s < 6; ++s) {  // Lb^64
    float nr = ar * ar - ai * ai;
    float ni = 2.f * ar * ai;
    ar = nr;
    ai = ni;
  }
  float cr = 0.f, ci = 0.f;
  for (int c = 0; c < NCHUNK; ++c) {
    carry_in[(size_t)c * P2_DIM + p]         = cr;
    carry_in[(size_t)c * P2_DIM + p + P_DIM] = ci;
    float tr = carry[(size_t)c * P2_DIM + p];
    float ti = carry[(size_t)c * P2_DIM + p + P_DIM];
    float nr = ar * cr - ai * ci + tr;
    float ni = ar * ci + ai * cr + ti;
    cr = nr;
    ci = ni;
  }
}

__global__ __launch_bounds__(P_DIM) void scanC_kernel(
    float* __restrict__ X, const float* __restrict__ carry_in,
    const float* __restrict__ lbr_a, const float* __restrict__ lbi_a) {
  int p = threadIdx.x;
  int c = blockIdx.x;
  if (c == 0) return;
  float cr = carry_in[(size_t)c * P2_DIM + p];
  float ci = carry_in[(size_t)c * P2_DIM + p + P_DIM];
  float lbr = lbr_a[p], lbi = lbi_a[p];
  float pr = lbr, pi = lbi;
  size_t base = (size_t)c * CHUNK * P2_DIM + p;
  for (int j = 0; j < CHUNK; ++j) {
    X[base]         += pr * cr - pi * ci;
    X[base + P_DIM] += pr * ci + pi * cr;
    float nr = pr * lbr - pi * lbi;
    float ni = pr * lbi + pi * lbr;
    pr = nr;
    pi = ni;
    base += P2_DIM;
  }
}

// ---------------------------------------------------------------------------
extern "C" void kernel_launch(void* const* d_in, const int* in_sizes, int n_in,
                              void* d_out, int out_size, void* d_ws, size_t ws_size,
                              hipStream_t stream) {
  const float* u       = (const float*)d_in[0];
  const float* Lre     = (const float*)d_in[1];
  const float* Lim     = (const float*)d_in[2];
  const float* Bin     = (const float*)d_in[3];
  const float* Cin     = (const float*)d_in[4];
  const float* D       = (const float*)d_in[5];
  const float* logstep = (const float*)d_in[6];

  float* ws = (float*)d_ws;
  float* lbr      = ws; ws += P_DIM;
  float* lbi      = ws; ws += P_DIM;
  float* scr      = ws; ws += P_DIM;
  float* sci      = ws; ws += P_DIM;
  float* carry    = ws; ws += (size_t)NCHUNK * P2_DIM;
  float* carry_in = ws; ws += (size_t)NCHUNK * P2_DIM;
  float* Bst      = ws; ws += (size_t)P2_DIM * H_DIM;
  float* Cst      = ws; ws += (size_t)H_DIM * P2_DIM;
  float* X        = ws; ws += (size_t)L_DIM * P2_DIM;  // Bu -> xs in-place

  pre_lambda_kernel<<<1, P_DIM, 0, stream>>>(Lre, Lim, logstep, lbr, lbi, scr, sci);
  build_B_kernel<<<(P_DIM * H_DIM) / 256, 256, 0, stream>>>(Bin, scr, sci, Bst);
  build_C_kernel<<<(H_DIM * P_DIM) / 256, 256, 0, stream>>>(Cin, Cst);

  // Bu = u (L,H) @ Bst(P2,H)^T -> X (L,P2)
  wmma_gemm_kernel<<<dim3(L_DIM / 64, P2_DIM / BN), 128, 0, stream>>>(
      u, Bst, X, H_DIM, P2_DIM, 0, nullptr, nullptr);

  scanA_kernel<<<NCHUNK, P_DIM, 0, stream>>>(X, lbr, lbi, carry);
  scanB_kernel<<<1, P_DIM, 0, stream>>>(carry, carry_in, lbr, lbi);
  scanC_kernel<<<NCHUNK, P_DIM, 0, stream>>>(X, carry_in, lbr, lbi);

  // y = xs (L,P2) @ Cst(H,P2)^T + D*u -> d_out (L,H)
  wmma_gemm_kernel<<<dim3(L_DIM / 64, H_DIM / BN), 128, 0, stream>>>(
      X, Cst, (float*)d_out, P2_DIM, H_DIM, 1, D, u);
}